// Composition_69372311765137
// MI455X (gfx1250) — compile-verified
//
#include <hip/hip_runtime.h>
#include <hip/hip_bf16.h>

#define NOBJ 64

// gfx1250 async global->LDS copy builtin operand types (from hipcc diagnostic):
//   param0: int4 addrspace(1)*  (global src, non-const)
//   param1: int4 addrspace(3)*  (LDS dst)
typedef int v4i __attribute__((ext_vector_type(4)));
typedef __attribute__((address_space(1))) v4i* gp4;
typedef __attribute__((address_space(3))) v4i* lp4;

struct F3 { float x, y, z; };  // 12-byte packed point (means layout [M,3])

__global__ __launch_bounds__(256) void Composition_69372311765137_kernel(
    const float* __restrict__ trans,    // [64,3]
    const float* __restrict__ rotors,   // [64,4]
    const float* __restrict__ means,    // [M,3]
    const float* __restrict__ quats,    // [M,4]  (w,x,y,z)
    const int*   __restrict__ indices,  // [M]
    float* __restrict__ out_means,      // [M,3]
    float* __restrict__ out_quats,      // [M,4]
    int M)
{
    // Object table staged in LDS: rotors as float4-aligned rows, trans packed.
    __shared__ __align__(16) float s_rot[NOBJ * 4];   // 1024 B
    __shared__ __align__(16) float s_trn[NOBJ * 3];   //  768 B

    const int t = threadIdx.x;

#if __has_builtin(__builtin_amdgcn_global_load_async_to_lds_b128) && \
    __has_builtin(__builtin_amdgcn_s_wait_asynccnt)
    // CDNA5 async DMA path: 64 lanes x 16B for rotors, 48 lanes x 16B for trans.
    if (t < 64) {
        __builtin_amdgcn_global_load_async_to_lds_b128(
            (gp4)(void*)(rotors + 4 * t), (lp4)(&s_rot[4 * t]), 0, 0);
    }
    if (t < 48) {
        __builtin_amdgcn_global_load_async_to_lds_b128(
            (gp4)(void*)(trans + 4 * t), (lp4)(&s_trn[4 * t]), 0, 0);
    }
    __builtin_amdgcn_s_wait_asynccnt(0);
#else
    // Fallback: cooperative VGPR-staged fill (blockDim.x == 256).
    if (t < NOBJ * 4) s_rot[t] = rotors[t];
    if (t < NOBJ * 3) s_trn[t] = trans[t];
#endif
    __syncthreads();

    const int i = blockIdx.x * 256 + t;
    if (i >= M) return;

    const int idx = indices[i] & (NOBJ - 1);

    // Streaming, coalesced wide loads: b128 for quats, b96 for means.
    const float4 qv = reinterpret_cast<const float4*>(quats)[i];
    const F3     m  = reinterpret_cast<const F3*>(means)[i];

    // One ds_load_b128 per lane for the rotor row.
    const float4 r  = reinterpret_cast<const float4*>(s_rot)[idx];
    const float rw = r.x, rx = r.y, ry = r.z, rz = r.w;

    // quat_rotate: u = 2*cross(r.xyz, v); out = v + rw*u + cross(r.xyz, u) + trans
    const float ux = 2.0f * (ry * m.z - rz * m.y);
    const float uy = 2.0f * (rz * m.x - rx * m.z);
    const float uz = 2.0f * (rx * m.y - ry * m.x);

    F3 om;
    om.x = m.x + rw * ux + (ry * uz - rz * uy) + s_trn[3 * idx + 0];
    om.y = m.y + rw * uy + (rz * ux - rx * uz) + s_trn[3 * idx + 1];
    om.z = m.z + rw * uz + (rx * uy - ry * ux) + s_trn[3 * idx + 2];

    // quat_mul(r, q), components stored (w,x,y,z)
    const float qw = qv.x, qx = qv.y, qy = qv.z, qz = qv.w;
    float4 oq;
    oq.x = rw * qw - rx * qx - ry * qy - rz * qz;
    oq.y = rw * qx + rx * qw + ry * qz - rz * qy;
    oq.z = rw * qy - rx * qz + ry * qw + rz * qx;
    oq.w = rw * qz + rx * qy - ry * qx + rz * qw;

    reinterpret_cast<F3*>(out_means)[i]     = om;   // b96 store
    reinterpret_cast<float4*>(out_quats)[i] = oq;   // b128 store
}

extern "C" void kernel_launch(void* const* d_in, const int* in_sizes, int n_in,
                              void* d_out, int out_size, void* d_ws, size_t ws_size,
                              hipStream_t stream) {
    const float* trans   = (const float*)d_in[0];
    const float* rotors  = (const float*)d_in[1];
    const float* means   = (const float*)d_in[2];
    const float* quats   = (const float*)d_in[3];
    const int*   indices = (const int*)d_in[4];
    const int M = in_sizes[4];                 // number of points

    float* out_means = (float*)d_out;          // first 3*M floats
    float* out_quats = out_means + (size_t)3 * (size_t)M;  // next 4*M floats

    const int block = 256;
    const int grid  = (M + block - 1) / block;
    Composition_69372311765137_kernel<<<grid, block, 0, stream>>>(
        trans, rotors, means, quats, indices, out_means, out_quats, M);
}